// SSL_73512660238656
// MI455X (gfx1250) — compile-verified
//
#include <hip/hip_runtime.h>
#include <hip/hip_bf16.h>

typedef __attribute__((ext_vector_type(2))) float v2f;
typedef __attribute__((ext_vector_type(8))) float v8f;

#define D 128
#define XS_STRIDE 132   // 128 + 4 pad: keeps float4 alignment, spreads LDS banks

// ---------------------------------------------------------------------------
// Degree / normalization kernels
// ---------------------------------------------------------------------------
__global__ void deg_init_k(float* deg, int n) {
    int i = blockIdx.x * blockDim.x + threadIdx.x;
    if (i < n) deg[i] = 1.0f;   // self-loop contributes +1
}

__global__ void deg_count_k(const long long* __restrict__ dst, float* deg, int e) {
    int i = blockIdx.x * blockDim.x + threadIdx.x;
    if (i < e) atomicAdd(&deg[(int)dst[i]], 1.0f);
}

__global__ void dinv_k(float* deg, int n) {
    int i = blockIdx.x * blockDim.x + threadIdx.x;
    if (i < n) deg[i] = rsqrtf(deg[i]);   // deg buffer now holds dinv
}

__global__ void edge_norm_k(const long long* __restrict__ src,
                            const long long* __restrict__ dst,
                            const float* __restrict__ dinv,
                            float* __restrict__ en, int e) {
    int i = blockIdx.x * blockDim.x + threadIdx.x;
    if (i < e) en[i] = dinv[(int)src[i]] * dinv[(int)dst[i]];
}

// ---------------------------------------------------------------------------
// WMMA GEMM: out[n,128] = X[n,128] @ W[128,128] + bscale * b
// block = 256 threads (8 waves); block computes 16 rows x 128 cols,
// wave w owns columns [16w, 16w+16). Uses V_WMMA_F32_16X16X4_F32 (exact fp32).
// ---------------------------------------------------------------------------
__global__ void __launch_bounds__(256)
gemm128_wmma_k(const float* __restrict__ X, const float* __restrict__ W,
               const float* __restrict__ bias, float bscale,
               float* __restrict__ out, int n) {
    __shared__ float xs[16 * XS_STRIDE];

    const int tid  = threadIdx.x;
    const int row0 = blockIdx.x * 16;
    const int wave = tid >> 5;
    const int lane = tid & 31;

    // ---- stage A strip (16x128) into LDS with float4 loads ----
    // 512 float4 slots, 256 threads -> 2 each
    for (int it = tid; it < 16 * 32; it += 256) {
        int r  = it >> 5;         // row in strip
        int c4 = it & 31;         // float4 column
        int gr = row0 + r;
        float4 v = make_float4(0.f, 0.f, 0.f, 0.f);
        if (gr < n) v = *(const float4*)(X + (size_t)gr * D + c4 * 4);
        *(float4*)(xs + r * XS_STRIDE + c4 * 4) = v;
    }
    __syncthreads();

    const int colg  = (wave << 4) + (lane & 15);   // global output column
    const int kpair = (lane >> 4) << 1;            // 0 or 2 within 4-wide K slab
    const int arow  = lane & 15;

    // ---- preload B fragments (W is small & L2 hot) ----
    v2f bf[32];
#pragma unroll
    for (int kk = 0; kk < 32; ++kk) {
        int kb = (kk << 2) + kpair;
        bf[kk].x = W[(size_t)kb * D + colg];
        bf[kk].y = W[(size_t)(kb + 1) * D + colg];
    }

    // ---- accumulator init with (scaled) bias ----
    float binit = bias ? bscale * bias[colg] : 0.0f;
    v8f c;
#pragma unroll
    for (int v = 0; v < 8; ++v) c[v] = binit;

    // ---- K loop: 32 x v_wmma_f32_16x16x4_f32 ----
#pragma unroll
    for (int kk = 0; kk < 32; ++kk) {
        int kb = (kk << 2) + kpair;
        v2f a;
        a.x = xs[arow * XS_STRIDE + kb];
        a.y = xs[arow * XS_STRIDE + kb + 1];
        c = __builtin_amdgcn_wmma_f32_16x16x4_f32(
                false, a, false, bf[kk], (short)0, c, false, false);
    }

    // ---- store D: lane holds col = lane&15, rows v + 8*(lane>>4) ----
    const int mbase = (lane >> 4) << 3;
#pragma unroll
    for (int v = 0; v < 8; ++v) {
        int gr = row0 + mbase + v;
        if (gr < n) out[(size_t)gr * D + colg] = c[v];
    }
}

// ---------------------------------------------------------------------------
// Elementwise / scatter kernels
// ---------------------------------------------------------------------------

// out = dinv[row]^2 * Y + b   (GCN self-loop + bias base, float4 per thread)
__global__ void gcn_base_k(const float* __restrict__ Y, const float* __restrict__ dinv,
                           const float* __restrict__ b, float* __restrict__ out, int n) {
    long long t = (long long)blockIdx.x * blockDim.x + threadIdx.x;
    if (t >= (long long)n * 32) return;
    int row = (int)(t >> 5);
    int c4  = (int)(t & 31);
    float s = dinv[row]; s *= s;
    float4 y  = *(const float4*)(Y + (size_t)row * D + c4 * 4);
    float4 bb = *(const float4*)(b + c4 * 4);
    float4 o = make_float4(s * y.x + bb.x, s * y.y + bb.y,
                           s * y.z + bb.z, s * y.w + bb.w);
    *(float4*)(out + (size_t)row * D + c4 * 4) = o;
}

// out[dst] += en[e] * Y[src]  — one wave per edge, float4 per lane
__global__ void gcn_scatter_k(const float* __restrict__ Y,
                              const long long* __restrict__ src,
                              const long long* __restrict__ dst,
                              const float* __restrict__ en,
                              float* __restrict__ out, int e) {
    long long t = (long long)blockIdx.x * blockDim.x + threadIdx.x;
    long long ei = t >> 5;
    if (ei >= e) return;
    int lane = (int)(t & 31);
    int s = (int)src[ei], d = (int)dst[ei];
    float w = en[ei];
    float4 v = *(const float4*)(Y + (size_t)s * D + lane * 4);
    float* o = out + (size_t)d * D + lane * 4;
    atomicAdd(o + 0, v.x * w);
    atomicAdd(o + 1, v.y * w);
    atomicAdd(o + 2, v.z * w);
    atomicAdd(o + 3, v.w * w);
}

// out[dst] += Y[src]  (GIN aggregation)
__global__ void gin_scatter_k(const float* __restrict__ Y,
                              const long long* __restrict__ src,
                              const long long* __restrict__ dst,
                              float* __restrict__ out, int e) {
    long long t = (long long)blockIdx.x * blockDim.x + threadIdx.x;
    long long ei = t >> 5;
    if (ei >= e) return;
    int lane = (int)(t & 31);
    int s = (int)src[ei], d = (int)dst[ei];
    float4 v = *(const float4*)(Y + (size_t)s * D + lane * 4);
    float* o = out + (size_t)d * D + lane * 4;
    atomicAdd(o + 0, v.x);
    atomicAdd(o + 1, v.y);
    atomicAdd(o + 2, v.z);
    atomicAdd(o + 3, v.w);
}

__global__ void relu_k(float* __restrict__ a, long long cnt4) {
    long long t = (long long)blockIdx.x * blockDim.x + threadIdx.x;
    if (t >= cnt4) return;
    float4 v = *(float4*)(a + t * 4);
    v.x = fmaxf(v.x, 0.f); v.y = fmaxf(v.y, 0.f);
    v.z = fmaxf(v.z, 0.f); v.w = fmaxf(v.w, 0.f);
    *(float4*)(a + t * 4) = v;
}

__global__ void copy_k(const float* __restrict__ s, float* __restrict__ d, long long cnt4) {
    long long t = (long long)blockIdx.x * blockDim.x + threadIdx.x;
    if (t >= cnt4) return;
    *(float4*)(d + t * 4) = *(const float4*)(s + t * 4);
}

__global__ void add_inplace_k(float* __restrict__ a, const float* __restrict__ b, long long cnt4) {
    long long t = (long long)blockIdx.x * blockDim.x + threadIdx.x;
    if (t >= cnt4) return;
    float4 x = *(float4*)(a + t * 4);
    float4 y = *(const float4*)(b + t * 4);
    x.x += y.x; x.y += y.y; x.z += y.z; x.w += y.w;
    *(float4*)(a + t * 4) = x;
}

// ---------------------------------------------------------------------------
// Host-side orchestration
// ---------------------------------------------------------------------------
extern "C" void kernel_launch(void* const* d_in, const int* in_sizes, int n_in,
                              void* d_out, int out_size, void* d_ws, size_t ws_size,
                              hipStream_t stream) {
    const float*     x      = (const float*)d_in[0];     // [n,128]
    const long long* ei     = (const long long*)d_in[1]; // [2,e] int64
    const float*     gcn_w1 = (const float*)d_in[2];
    const float*     gcn_b1 = (const float*)d_in[3];
    const float*     gcn_w2 = (const float*)d_in[4];
    const float*     gcn_b2 = (const float*)d_in[5];
    const float*     gin_w1 = (const float*)d_in[6];
    const float*     gin_b1 = (const float*)d_in[7];
    const float*     gin_w2 = (const float*)d_in[8];
    const float*     gin_b2 = (const float*)d_in[9];
    const float*     lin_w  = (const float*)d_in[10];
    const float*     lin_b  = (const float*)d_in[11];
    const float*     fc_w   = (const float*)d_in[12];
    const float*     fc_b   = (const float*)d_in[13];

    const int n = in_sizes[0] / D;
    const int e = in_sizes[1] / 2;
    const long long* src = ei;
    const long long* dst = ei + e;

    // workspace layout (16B aligned chunks)
    float* deg = (float*)d_ws;                       // n   (becomes dinv)
    float* en  = deg + ((n + 3) & ~3);               // e
    float* A   = en + ((e + 3) & ~3);                // n*128
    float* C   = A + (size_t)n * D;                  // n*128
    float* B   = (float*)d_out;                      // n*128 (x_gcn lives here)

    const long long cnt4 = (long long)n * (D / 4);
    const int TB = 256;
    auto blk  = [&](long long cnt) { return (unsigned)((cnt + TB - 1) / TB); };
    const unsigned gN   = blk(n);
    const unsigned gE   = blk(e);
    const unsigned gNE  = blk((long long)n * 32);
    const unsigned gEW  = blk((long long)e * 32);
    const unsigned g4   = blk(cnt4);
    const unsigned gGemm = (unsigned)((n + 15) / 16);

    // --- normalization ---
    deg_init_k<<<gN, TB, 0, stream>>>(deg, n);
    deg_count_k<<<gE, TB, 0, stream>>>(dst, deg, e);
    dinv_k<<<gN, TB, 0, stream>>>(deg, n);
    edge_norm_k<<<gE, TB, 0, stream>>>(src, dst, deg, en, e);

    // --- GCN branch ---
    gemm128_wmma_k<<<gGemm, TB, 0, stream>>>(x, gcn_w1, nullptr, 0.f, A, n);   // A = Y1
    gcn_base_k<<<gNE, TB, 0, stream>>>(A, deg, gcn_b1, C, n);                  // C = dinv^2*Y1 + b1
    gcn_scatter_k<<<gEW, TB, 0, stream>>>(A, src, dst, en, C, e);              // C += scatter
    relu_k<<<g4, TB, 0, stream>>>(C, cnt4);                                    // C = H
    gemm128_wmma_k<<<gGemm, TB, 0, stream>>>(C, gcn_w2, nullptr, 0.f, A, n);   // A = Y2
    gcn_base_k<<<gNE, TB, 0, stream>>>(A, deg, gcn_b2, B, n);                  // B = dinv^2*Y2 + b2
    gcn_scatter_k<<<gEW, TB, 0, stream>>>(A, src, dst, en, B, e);              // B = x_gcn

    // --- GIN branch ---
    copy_k<<<g4, TB, 0, stream>>>(x, C, cnt4);                                 // C = X
    gin_scatter_k<<<gEW, TB, 0, stream>>>(x, src, dst, C, e);                  // C = X + agg(X)
    gemm128_wmma_k<<<gGemm, TB, 0, stream>>>(C, gin_w1, gin_b1, 1.f, A, n);    // A = G
    copy_k<<<g4, TB, 0, stream>>>(A, C, cnt4);                                 // C = G
    gin_scatter_k<<<gEW, TB, 0, stream>>>(A, src, dst, C, e);                  // C = G + agg(G)
    gemm128_wmma_k<<<gGemm, TB, 0, stream>>>(C, gin_w2, gin_b2, 1.f, A, n);    // A = x_gin

    // --- fusion: out = ((x_gin + x_gcn) @ lin_w + 2*lin_b) @ fc_w + fc_b ---
    add_inplace_k<<<g4, TB, 0, stream>>>(A, B, cnt4);                          // A = x_gin + x_gcn
    gemm128_wmma_k<<<gGemm, TB, 0, stream>>>(A, lin_w, lin_b, 2.f, C, n);      // C = U
    gemm128_wmma_k<<<gGemm, TB, 0, stream>>>(C, fc_w, fc_b, 1.f, B, n);        // d_out
}